// SAGEConv_7181185319698
// MI455X (gfx1250) — compile-verified
//
#include <hip/hip_runtime.h>

typedef __attribute__((ext_vector_type(2))) float v2f;
typedef __attribute__((ext_vector_type(8))) float v8f;

#define D 128          // D_IN == D_OUT == 128
#define ROWS_PER_BLK 64
#define LDS_STRIDE 132 // 128 + 4 pad: 132 mod 64 = 4 -> conflict-free fragment reads,
                       // row base stays 16B aligned (132*4 = 528 = 33*16)

// ---------------------------------------------------------------------------
// Kernel 1: zero the aggregation buffer and degree buffer in workspace.
// (harness poisons d_ws with 0xAA; we must re-zero every call)
// ---------------------------------------------------------------------------
__global__ void sage_zero_ws(float* __restrict__ agg, float* __restrict__ deg,
                             int n_nodes) {
    long long stride = (long long)gridDim.x * blockDim.x;
    long long i0 = (long long)blockIdx.x * blockDim.x + threadIdx.x;
    long long tot = (long long)n_nodes * D;
    for (long long i = i0; i < tot; i += stride) agg[i] = 0.0f;
    for (long long i = i0; i < n_nodes; i += stride) deg[i] = 0.0f;
}

// ---------------------------------------------------------------------------
// Kernel 2: edge scatter. One wave (32 lanes) per edge:
//   lanes cooperatively read the 128-float source row as float4 (global_load_b128)
//   and scatter-add it into agg[dst] with global_atomic_add_f32.
// inputs/agg/edge-lists all fit in the 192MB L2, so this phase is L2-resident.
// ---------------------------------------------------------------------------
__global__ void __launch_bounds__(256)
sage_scatter(const float* __restrict__ inputs,
             const int* __restrict__ src,
             const int* __restrict__ dst,
             float* __restrict__ agg,
             float* __restrict__ deg,
             int n_edges) {
    const int lane  = threadIdx.x & 31;
    const int wave  = (blockIdx.x * blockDim.x + threadIdx.x) >> 5;
    const int nwave = (gridDim.x * blockDim.x) >> 5;

    for (int e = wave; e < n_edges; e += nwave) {
        const int s = src[e];
        const int d = dst[e];
        // 32 lanes x float4 = 128 floats = one full row
        const float4 v = ((const float4*)(inputs + (long long)s * D))[lane];
        float* o = agg + (long long)d * D + lane * 4;
        atomicAdd(o + 0, v.x);
        atomicAdd(o + 1, v.y);
        atomicAdd(o + 2, v.z);
        atomicAdd(o + 3, v.w);
        if (lane == 0) atomicAdd(deg + d, 1.0f);
    }
}

// ---------------------------------------------------------------------------
// Kernel 3: fused dual-GEMM + mean-normalize + bias + ReLU with
// V_WMMA_F32_16X16X4_F32 (A 16x4, B 4x16, C/D 16x16 f32).
//
// Block tile: 64 rows x 128 cols, 8 waves. Wave w owns cols [16w,16w+16) and
// FOUR 16-row sub-tiles (4 x v8f accumulators), so each B fragment is loaded
// once per k-step and reused 4x. A (self + normalized neighbor) is staged in
// LDS once per block — shared by all 8 waves instead of re-read 8x from L0.
//
// Fragment layouts (ISA 7.12.2, 32-bit):
//   A 16x4 : lanes 0-15 hold M=lane,    K=k+0 (v0), k+1 (v1)
//            lanes 16-31 hold M=lane-16, K=k+2 (v0), k+3 (v1)
//   B 4x16 : same split over N
//   C/D    : N = lane&15 ; VGPR r -> M = r + (lane>=16 ? 8 : 0)
// ---------------------------------------------------------------------------
__global__ void __launch_bounds__(256)
sage_wmma_gemm(const float* __restrict__ inputs,
               const float* __restrict__ agg,
               const float* __restrict__ deg,
               const float* __restrict__ W_self,
               const float* __restrict__ W_neigh,
               const float* __restrict__ bias,
               float* __restrict__ out,
               int n_nodes) {
    extern __shared__ float smem[];
    float* As   = smem;                               // [64][LDS_STRIDE]
    float* An   = smem + ROWS_PER_BLK * LDS_STRIDE;   // [64][LDS_STRIDE]
    float* sInv = smem + 2 * ROWS_PER_BLK * LDS_STRIDE; // [64]

    const int t       = threadIdx.x;
    const int lane    = t & 31;
    const int waveId  = t >> 5;                 // 0..7 -> column tile
    const int rowBase = blockIdx.x * ROWS_PER_BLK;

    // --- stage 1: per-row 1/max(deg,1) via v_rcp_f32 (deg is an int count;
    //     rcp precision is ample) ---
    if (t < ROWS_PER_BLK) {
        int r = rowBase + t;
        if (r >= n_nodes) r = n_nodes - 1;
        sInv[t] = __builtin_amdgcn_rcpf(fmaxf(deg[r], 1.0f));
    }
    __syncthreads();

    // --- stage 2: cooperative LDS staging of A_self and normalized A_neigh.
    //     64 rows x 32 float4 = 2048 b128 transfers, 8 per thread. ---
    for (int j = t; j < ROWS_PER_BLK * (D / 4); j += 256) {
        const int r  = j >> 5;          // row in tile
        const int c4 = j & 31;          // float4 column
        int rg = rowBase + r;
        if (rg >= n_nodes) rg = n_nodes - 1;
        const float4 vs = ((const float4*)(inputs + (long long)rg * D))[c4];
        const float4 vn = ((const float4*)(agg    + (long long)rg * D))[c4];
        const float  sc = sInv[r];
        *(float4*)(As + r * LDS_STRIDE + c4 * 4) = vs;
        float4 vnn = make_float4(vn.x * sc, vn.y * sc, vn.z * sc, vn.w * sc);
        *(float4*)(An + r * LDS_STRIDE + c4 * 4) = vnn;
    }
    __syncthreads();

    // --- stage 3: K-loop with 8 WMMAs per step (4 row-tiles x 2 matrices) ---
    const int half = lane >> 4;                 // 0 | 1
    const int koff = half * 2;                  // K sub-offset per layout
    const int aRow = lane & 15;                 // row within a 16-row sub-tile
    const int nB   = waveId * 16 + (lane & 15); // B/C column for this lane

    v8f acc0 = {0.f,0.f,0.f,0.f,0.f,0.f,0.f,0.f};
    v8f acc1 = acc0, acc2 = acc0, acc3 = acc0;

    #pragma unroll
    for (int k = 0; k < D; k += 4) {
        v2f b_s, b_n;
        b_s[0] = W_self [(k + koff + 0) * D + nB];
        b_s[1] = W_self [(k + koff + 1) * D + nB];
        b_n[0] = W_neigh[(k + koff + 0) * D + nB];
        b_n[1] = W_neigh[(k + koff + 1) * D + nB];

        #pragma unroll
        for (int rt = 0; rt < 4; ++rt) {
            const float* pS = As + (rt * 16 + aRow) * LDS_STRIDE + k + koff;
            const float* pN = An + (rt * 16 + aRow) * LDS_STRIDE + k + koff;
            v2f a_s; a_s[0] = pS[0]; a_s[1] = pS[1];   // ds_load_b64
            v2f a_n; a_n[0] = pN[0]; a_n[1] = pN[1];   // ds_load_b64
            v8f* accp = rt == 0 ? &acc0 : rt == 1 ? &acc1 : rt == 2 ? &acc2 : &acc3;
            *accp = __builtin_amdgcn_wmma_f32_16x16x4_f32(
                false, a_s, false, b_s, (short)0, *accp, false, false);
            *accp = __builtin_amdgcn_wmma_f32_16x16x4_f32(
                false, a_n, false, b_n, (short)0, *accp, false, false);
        }
    }

    // --- epilogue: bias + ReLU, store 4 tiles per C/D layout ---
    const float bv = bias[nB];
    #pragma unroll
    for (int rt = 0; rt < 4; ++rt) {
        const v8f& a = rt == 0 ? acc0 : rt == 1 ? acc1 : rt == 2 ? acc2 : acc3;
        #pragma unroll
        for (int r = 0; r < 8; ++r) {
            const int m = rowBase + rt * 16 + r + half * 8;
            if (m < n_nodes) {
                const float v = a[r] + bv;
                out[(long long)m * D + nB] = v > 0.0f ? v : 0.0f;
            }
        }
    }
}

// ---------------------------------------------------------------------------
// Launch: zero ws -> edge scatter -> fused WMMA GEMM epilogue.
// d_in order: inputs, W_self, W_neigh, b, src, dst
// d_ws layout: [agg: n_nodes*128 f32][deg: n_nodes f32]
// ---------------------------------------------------------------------------
extern "C" void kernel_launch(void* const* d_in, const int* in_sizes, int n_in,
                              void* d_out, int out_size, void* d_ws, size_t ws_size,
                              hipStream_t stream) {
    const float* inputs  = (const float*)d_in[0];
    const float* W_self  = (const float*)d_in[1];
    const float* W_neigh = (const float*)d_in[2];
    const float* bias    = (const float*)d_in[3];
    const int*   src     = (const int*)d_in[4];
    const int*   dst     = (const int*)d_in[5];
    float*       out     = (float*)d_out;

    const int n_nodes = in_sizes[0] / D;
    const int n_edges = in_sizes[4];

    float* agg = (float*)d_ws;
    float* deg = agg + (size_t)n_nodes * D;

    // 1) zero workspace
    sage_zero_ws<<<2048, 256, 0, stream>>>(agg, deg, n_nodes);

    // 2) scatter-add edges (one wave per edge, grid-stride)
    sage_scatter<<<4096, 256, 0, stream>>>(inputs, src, dst, agg, deg, n_edges);

    // 3) fused normalize + dual WMMA GEMM + bias + ReLU
    const int blocks = (n_nodes + ROWS_PER_BLK - 1) / ROWS_PER_BLK;  // 1563
    const size_t smemBytes =
        (size_t)(2 * ROWS_PER_BLK * LDS_STRIDE + ROWS_PER_BLK) * sizeof(float);
    sage_wmma_gemm<<<blocks, 256, smemBytes, stream>>>(inputs, agg, deg,
                                                       W_self, W_neigh, bias,
                                                       out, n_nodes);
}